// Attention_86131274154182
// MI455X (gfx1250) — compile-verified
//
#include <hip/hip_runtime.h>
#include <hip/hip_bf16.h>

#define BS 4
#define QL 16
#define DIM 4096
#define NH 32
#define HD 128
#define CL 4096
#define TT (CL + QL)   // 4112

typedef float v2f __attribute__((ext_vector_type(2)));
typedef float v8f __attribute__((ext_vector_type(8)));

#define GEMM_BK 64
#define LPAD 68   // LDS row pitch in floats: multiple of 4 (16B float4 stores), spreads banks

// CDNA5 async global->LDS path (ASYNCcnt), guarded so the build can't regress.
#if defined(__has_builtin)
#if __has_builtin(__builtin_amdgcn_global_load_async_to_lds_b128) && \
    __has_builtin(__builtin_amdgcn_s_wait_asynccnt)
#define USE_ASYNC 1
#endif
#endif
#ifndef USE_ASYNC
#define USE_ASYNC 0
#endif

#if USE_ASYNC
typedef int v4i_vs __attribute__((vector_size(4 * sizeof(int))));
typedef __attribute__((address_space(1))) v4i_vs* as1_v4i;
typedef __attribute__((address_space(3))) v4i_vs* as3_v4i;
#endif

// ---------------------------------------------------------------------------
// Y[m,n] = sum_k A[m,k] * W[n,k]   (A: [64,4096] row-major, W: [4096,4096] row-major)
// One block: 16 (M) x 128 (N) output tile, 8 waves = 8 column sub-tiles of 16.
// Double-buffered K loop; weight tiles streamed straight into LDS with
// GLOBAL_LOAD_ASYNC_TO_LDS_B128 (no VGPR staging), fenced by s_wait_asynccnt.
// ---------------------------------------------------------------------------
__global__ __launch_bounds__(256) void gemm_xwT_kernel(
    const float* __restrict__ A,
    const float* __restrict__ W0, const float* __restrict__ W1, const float* __restrict__ W2,
    float* __restrict__ O0, float* __restrict__ O1, float* __restrict__ O2)
{
    extern __shared__ float gsm[];   // 2 * (16+128) * LPAD floats

    const int z = blockIdx.z;
    const float* __restrict__ W = (z == 0) ? W0 : (z == 1) ? W1 : W2;
    float* __restrict__ O       = (z == 0) ? O0 : (z == 1) ? O1 : O2;

    const int nb = blockIdx.x * 128;
    const int mb = blockIdx.y * 16;
    const int t  = threadIdx.x;
    const int wv = t >> 5;          // wave 0..7
    const int lane = t & 31;
    const int lm = lane & 15;       // fragment row/col index
    const int hi = lane >> 4;       // K half select
    const int ns = wv * 16;

    const int BUF = (16 + 128) * LPAD;

    const int rowA = t >> 4;
    const int c4A  = (t & 15) * 4;

    v8f acc = {0.f, 0.f, 0.f, 0.f, 0.f, 0.f, 0.f, 0.f};

    auto issue = [&](int k0, int buf) {
        float* As = gsm + buf * BUF;
        float* Bs = As + 16 * LPAD;
        // A tile: 16 x 64 floats = 256 x b128, 1 per thread (coalesced)
        {
            const float* gA = &A[(size_t)(mb + rowA) * DIM + k0 + c4A];
            float* lA = &As[rowA * LPAD + c4A];
#if USE_ASYNC
            __builtin_amdgcn_global_load_async_to_lds_b128((as1_v4i)gA, (as3_v4i)lA, 0, 0);
#else
            *(float4*)lA = *(const float4*)gA;
#endif
        }
        // B tile: 128 x 64 floats = 2048 x b128, 8 per thread (coalesced)
        #pragma unroll
        for (int i = 0; i < 8; ++i) {
            const int idx = t + i * 256;
            const int row = idx >> 4;
            const int c4  = (idx & 15) * 4;
            const float* gB = &W[(size_t)(nb + row) * DIM + k0 + c4];
            float* lB = &Bs[row * LPAD + c4];
#if USE_ASYNC
            __builtin_amdgcn_global_load_async_to_lds_b128((as1_v4i)gB, (as3_v4i)lB, 0, 0);
#else
            *(float4*)lB = *(const float4*)gB;
#endif
        }
    };

    issue(0, 0);

    const int NC = DIM / GEMM_BK;   // 64 chunks
    for (int c = 0; c < NC; ++c) {
        const int cur = c & 1;
#if USE_ASYNC
        __builtin_amdgcn_s_wait_asynccnt(0);   // chunk c landed in LDS
#endif
        __syncthreads();
        if (c + 1 < NC) issue((c + 1) * GEMM_BK, cur ^ 1);  // prefetch next chunk

        const float* As = gsm + cur * BUF;
        const float* Bs = As + 16 * LPAD;
        #pragma unroll
        for (int kk = 0; kk < GEMM_BK; kk += 4) {
            // A frag: lane=M, V0/V1 = K(2*hi), K(2*hi+1)  -> one float2
            v2f a = *(const v2f*)&As[lm * LPAD + kk + 2 * hi];
            // B frag: lane=N, same K split               -> one float2
            v2f b = *(const v2f*)&Bs[(ns + lm) * LPAD + kk + 2 * hi];
            acc = __builtin_amdgcn_wmma_f32_16x16x4_f32(false, a, false, b,
                                                        (short)0, acc, false, false);
        }
        __syncthreads();   // chunk c fully consumed before its buffer is re-filled
    }

    // D layout: lane column = nb+ns+lm; VGPR j -> row mb + j + 8*hi
    #pragma unroll
    for (int j = 0; j < 8; ++j) {
        O[(size_t)(mb + j + 8 * hi) * DIM + nb + ns + lm] = acc[j];
    }
}

// ---------------------------------------------------------------------------
// RoPE q,k + scatter rope'd k and raw v into tail (t = CL..CL+15) of k_new/v_new.
// One thread per even/odd pair; 64 rows * 2048 pairs = 131072 threads.
// ---------------------------------------------------------------------------
__global__ __launch_bounds__(256) void rope_scatter_kernel(
    const float* __restrict__ xq, const float* __restrict__ xk, const float* __restrict__ xv,
    const float* __restrict__ fcos, const float* __restrict__ fsin,
    float* __restrict__ xq_r, float* __restrict__ k_out, float* __restrict__ v_out)
{
    const int idx = blockIdx.x * 256 + threadIdx.x;
    const int row = idx >> 11;      // b*16+q
    const int p   = idx & 2047;     // pair within row
    const int h   = p >> 6;
    const int pd  = p & 63;
    const int q   = row & 15;
    const int b   = row >> 4;
    const int col = h * HD + 2 * pd;

    const float c = fcos[q * (HD / 2) + pd];
    const float s = fsin[q * (HD / 2) + pd];

    const size_t src = (size_t)row * DIM + col;
    // q -> workspace (rope'd)
    {
        const float x0 = xq[src], x1 = xq[src + 1];
        xq_r[src]     = x0 * c - x1 * s;
        xq_r[src + 1] = x0 * s + x1 * c;
    }
    const size_t kd = (size_t)b * TT * DIM + (size_t)(CL + q) * DIM + col;
    // k -> k_new tail (rope'd)
    {
        const float x0 = xk[src], x1 = xk[src + 1];
        k_out[kd]     = x0 * c - x1 * s;
        k_out[kd + 1] = x0 * s + x1 * c;
    }
    // v -> v_new tail (copy)
    v_out[kd]     = xv[src];
    v_out[kd + 1] = xv[src + 1];
}

// ---------------------------------------------------------------------------
// Flash attention, one block per (b,h). 8 waves take interleaved 16-key tiles,
// each keeps private online-softmax state + O[16x128] accumulator in VGPRs.
// Cross-wave merge through LDS at the end.
// ---------------------------------------------------------------------------
__global__ __launch_bounds__(256) void flash_attn_kernel(
    const float* __restrict__ Qr,   // [64, 4096] rope'd queries
    const float* __restrict__ Kn,   // [4, 4112, 32, 128]
    const float* __restrict__ Vn,   // [4, 4112, 32, 128]
    const float* __restrict__ mask, // [16, 4112]
    float* __restrict__ AO)         // [64, 4096]
{
    extern __shared__ float smem[];
    float* sQ  = smem;                       // 16 * 132
    float* sP  = sQ + 16 * 132;              // 8 * 16 * 20
    float* sO  = sP + 8 * 16 * 20;           // 8 * 16 * 128
    float* sM  = sO + 8 * QL * HD;           // 8 * 16
    float* sL  = sM + 8 * QL;                // 8 * 16
    float* sMM = sL + 8 * QL;                // 16
    float* sLL = sMM + QL;                   // 16

    const int bh = blockIdx.x;
    const int b = bh >> 5, h = bh & 31;
    const int t = threadIdx.x;
    const int wv = t >> 5, lane = t & 31, lm = lane & 15, hi = lane >> 4;

    const float* __restrict__ Qp = Qr + (size_t)b * QL * DIM + (size_t)h * HD;
    const float* __restrict__ Kp = Kn + (size_t)b * TT * DIM + (size_t)h * HD;
    const float* __restrict__ Vp = Vn + (size_t)b * TT * DIM + (size_t)h * HD;

    // Stage Q (16x128) into padded LDS: 512 float4, 2 per thread
    #pragma unroll
    for (int i = 0; i < 2; ++i) {
        const int idx = t + i * 256;
        const int row = idx >> 5;
        const int c4  = (idx & 31) * 4;
        *(float4*)&sQ[row * 132 + c4] = *(const float4*)&Qp[(size_t)row * DIM + c4];
    }
    __syncthreads();

    const float inv_sqrt = 0.08838834764831845f; // 1/sqrt(128)
    float run_m[8], run_l[8];
    v8f Oacc[8];
    #pragma unroll
    for (int j = 0; j < 8; ++j) {
        run_m[j] = -1e30f;
        run_l[j] = 0.f;
        Oacc[j] = v8f{0.f, 0.f, 0.f, 0.f, 0.f, 0.f, 0.f, 0.f};
    }

    float* sPw = sP + wv * 16 * 20;

    for (int t0 = wv * 16; t0 < TT; t0 += 128) {   // wave-uniform trip count
        // ---- S = Q K^T tile [16 queries x 16 keys], K-dim = 128
        v8f sacc = {0.f, 0.f, 0.f, 0.f, 0.f, 0.f, 0.f, 0.f};
        #pragma unroll
        for (int d0 = 0; d0 < HD; d0 += 4) {
            v2f a  = *(const v2f*)&sQ[lm * 132 + d0 + 2 * hi];
            v2f bk = *(const v2f*)&Kp[(size_t)(t0 + lm) * DIM + d0 + 2 * hi];
            sacc = __builtin_amdgcn_wmma_f32_16x16x4_f32(false, a, false, bk,
                                                         (short)0, sacc, false, false);
        }
        // ---- online softmax update (row = query in VGPR index, col = key in lane)
        float alpha[8];
        #pragma unroll
        for (int j = 0; j < 8; ++j) {
            const int m = j + 8 * hi;
            float sv = sacc[j] * inv_sqrt + mask[m * TT + (t0 + lm)];
            float mx = sv;
            mx = fmaxf(mx, __shfl_xor(mx, 1, 32));
            mx = fmaxf(mx, __shfl_xor(mx, 2, 32));
            mx = fmaxf(mx, __shfl_xor(mx, 4, 32));
            mx = fmaxf(mx, __shfl_xor(mx, 8, 32));
            const float nm = fmaxf(run_m[j], mx);
            const float pv = __expf(sv - nm);
            float sm = pv;
            sm += __shfl_xor(sm, 1, 32);
            sm += __shfl_xor(sm, 2, 32);
            sm += __shfl_xor(sm, 4, 32);
            sm += __shfl_xor(sm, 8, 32);
            const float al = __expf(run_m[j] - nm);
            run_l[j] = run_l[j] * al + sm;
            run_m[j] = nm;
            alpha[j] = al;
            sPw[m * 20 + lm] = pv;   // D-layout -> A-layout transpose via LDS
        }
        // ---- rescale running O
        #pragma unroll
        for (int c = 0; c < 8; ++c)
            #pragma unroll
            for (int j = 0; j < 8; ++j)
                Oacc[c][j] *= alpha[j];
        // ---- O += P * V  (K-dim = 16 keys, 8 column chunks of head_dim)
        #pragma unroll
        for (int c = 0; c < 8; ++c) {
            const int n0 = c * 16;
            #pragma unroll
            for (int s = 0; s < 4; ++s) {
                const int kt = s * 4 + 2 * hi;
                v2f aP = *(const v2f*)&sPw[lm * 20 + s * 4 + 2 * hi];
                v2f bv;
                bv.x = Vp[(size_t)(t0 + kt) * DIM + n0 + lm];
                bv.y = Vp[(size_t)(t0 + kt + 1) * DIM + n0 + lm];
                Oacc[c] = __builtin_amdgcn_wmma_f32_16x16x4_f32(false, aP, false, bv,
                                                                (short)0, Oacc[c], false, false);
            }
        }
    }

    // ---- spill per-wave state, merge across waves
    #pragma unroll
    for (int c = 0; c < 8; ++c)
        #pragma unroll
        for (int j = 0; j < 8; ++j)
            sO[wv * (QL * HD) + (j + 8 * hi) * HD + c * 16 + lm] = Oacc[c][j];
    if (lm == 0) {
        #pragma unroll
        for (int j = 0; j < 8; ++j) {
            sM[wv * QL + j + 8 * hi] = run_m[j];
            sL[wv * QL + j + 8 * hi] = run_l[j];
        }
    }
    __syncthreads();
    if (t < QL) {
        float M = -1e30f;
        for (int w = 0; w < 8; ++w) M = fmaxf(M, sM[w * QL + t]);
        float L = 0.f;
        for (int w = 0; w < 8; ++w) L += sL[w * QL + t] * __expf(sM[w * QL + t] - M);
        sMM[t] = M;
        sLL[t] = L;
    }
    __syncthreads();
    for (int idx = t; idx < QL * HD; idx += 256) {
        const int q = idx >> 7, d = idx & 127;
        const float M = sMM[q];
        float acc = 0.f;
        #pragma unroll
        for (int w = 0; w < 8; ++w)
            acc += sO[w * (QL * HD) + q * HD + d] * __expf(sM[w * QL + q] - M);
        AO[(size_t)(b * QL + q) * DIM + h * HD + d] = acc / sLL[q];
    }
}

// ---------------------------------------------------------------------------
extern "C" void kernel_launch(void* const* d_in, const int* in_sizes, int n_in,
                              void* d_out, int out_size, void* d_ws, size_t ws_size,
                              hipStream_t stream) {
    const float* x    = (const float*)d_in[0];
    const float* kc   = (const float*)d_in[1];
    const float* vc   = (const float*)d_in[2];
    const float* fcos = (const float*)d_in[3];
    const float* fsin = (const float*)d_in[4];
    const float* mask = (const float*)d_in[5];
    const float* wq   = (const float*)d_in[6];
    const float* wk   = (const float*)d_in[7];
    const float* wvp  = (const float*)d_in[8];
    const float* wo   = (const float*)d_in[9];

    float* out   = (float*)d_out;
    float* k_new = out;
    float* v_new = out + (size_t)BS * TT * DIM;
    float* y     = out + 2 * (size_t)BS * TT * DIM;

    float* ws  = (float*)d_ws;
    float* xq  = ws;                 // [64,4096]
    float* xk  = ws + 262144;        // [64,4096]
    float* xv  = ws + 524288;        // [64,4096]
    float* xqr = ws + 786432;        // [64,4096] rope'd q
    float* ao  = ws + 1048576;       // [64,4096] attention output

    // 1) cache -> output concat (t < CL), per-batch contiguous chunks
    for (int b = 0; b < BS; ++b) {
        (void)hipMemcpyAsync(k_new + (size_t)b * TT * DIM, kc + (size_t)b * CL * DIM,
                             (size_t)CL * DIM * sizeof(float), hipMemcpyDeviceToDevice, stream);
        (void)hipMemcpyAsync(v_new + (size_t)b * TT * DIM, vc + (size_t)b * CL * DIM,
                             (size_t)CL * DIM * sizeof(float), hipMemcpyDeviceToDevice, stream);
    }

    const size_t gemm_smem = 2 * (16 + 128) * LPAD * sizeof(float);  // 78,336 B

    // 2) QKV projections (WMMA f32, async double-buffered weight streaming)
    dim3 g1(DIM / 128, (BS * QL) / 16, 3);
    gemm_xwT_kernel<<<g1, 256, gemm_smem, stream>>>(x, wq, wk, wvp, xq, xk, xv);

    // 3) RoPE + tail scatter
    rope_scatter_kernel<<<(BS * QL * DIM / 2) / 256, 256, 0, stream>>>(
        xq, xk, xv, fcos, fsin, xqr, k_new, v_new);

    // 4) flash attention (one block per (b,h)); dynamic LDS ~85 KB
    const size_t smem_bytes =
        (16 * 132 + 8 * 16 * 20 + 8 * QL * HD + 8 * QL + 8 * QL + QL + QL) * sizeof(float);
    flash_attn_kernel<<<BS * NH, 256, smem_bytes, stream>>>(xqr, k_new, v_new, mask, ao);

    // 5) output projection -> d_out
    dim3 g2(DIM / 128, (BS * QL) / 16, 1);
    gemm_xwT_kernel<<<g2, 256, gemm_smem, stream>>>(ao, wo, wo, wo, y, y, y);
}